// DCTLayer_80144089743494
// MI455X (gfx1250) — compile-verified
//
#include <hip/hip_runtime.h>
#include <math.h>

// ---------------------------------------------------------------------------
// 2-D DCT-II (B=128, H=W=256, C=3) + log-abs-normalize.
// Two batched WMMA f32 GEMMs (V_WMMA_F32_16X16X4_F32, wave32) with the fixed
// DCT matrix D. B panels are staged to LDS with the Tensor Data Mover
// (tensor_load_to_lds, double-buffered, s_wait_tensorcnt) and consumed with
// ds_loads; each block covers 4 M-tiles so the panel is reused 4x.
//
//   stage1: Tt[b,c,w,h'] = sum_h D[h',h] * x[b,h,w,c]
//   stage2: out[b,h',k,c] = (log(|sum_w D[k,w]*Tt[b,c,w,h']|+eps)-mean)/std
//
// ws layout:  Dm : 65536 floats (256 KB);  Tt : 128*3*256*256 floats (~101 MB)
// ---------------------------------------------------------------------------

typedef float v2f __attribute__((ext_vector_type(2)));
typedef float v4f __attribute__((ext_vector_type(4)));
typedef float v8f __attribute__((ext_vector_type(8)));
typedef unsigned int u32x4 __attribute__((ext_vector_type(4)));
typedef int i32x4 __attribute__((ext_vector_type(4)));
typedef int i32x8 __attribute__((ext_vector_type(8)));

#define WMMA_F32(a, b, c) \
  __builtin_amdgcn_wmma_f32_16x16x4_f32(false, (a), false, (b), (short)0, (c), false, false)

#ifndef M_PI
#define M_PI 3.14159265358979323846
#endif

#if defined(__has_builtin)
#if __has_builtin(__builtin_amdgcn_tensor_load_to_lds) && \
    __has_builtin(__builtin_amdgcn_s_wait_tensorcnt)
#define HAVE_TDM 1
#endif
#endif
#ifndef HAVE_TDM
#define HAVE_TDM 0
#endif

// LDS panel: 32 K-rows x 64 N-cols, padded to 72-float row stride so that the
// two lane-halves (rows r and r+2) of each ds_load hit disjoint bank ranges.
#define PANEL_ROWS   32
#define PANEL_COLS   64
#define PANEL_STRIDE 72

#if HAVE_TDM
// 2-D TDM load: PANEL_ROWS x PANEL_COLS f32 tile (row stride = stride_elems)
// from global into LDS, with 8-DWORD padding every 64 DWORDs (-> 72f rows).
// 6-arg builtin form: (g0 u32x4, g1 i32x8, g2 i32x4, g3 i32x4, i32x8, cpol).
__device__ __forceinline__ void tdm_load_2d(unsigned lds_byte_addr,
                                            const float* gsrc,
                                            unsigned stride_elems) {
  unsigned long long ga = (unsigned long long)(const void*)gsrc;
  u32x4 g0;
  g0[0] = 1u;                                   // count=1, user desc, no gather
  g0[1] = lds_byte_addr;                        // LDS byte address
  g0[2] = (unsigned)ga;                         // global_addr[31:0]
  g0[3] = (unsigned)(ga >> 32) | (2u << 30);    // global_addr[56:32] | type=2
  i32x8 g1;
  g1[0] = (int)((2u << 16)                      // data_size = 4B
              | (1u << 20)                      // pad_enable
              | (5u << 22)                      // pad_interval: 64 DWORDs
              | (7u << 25));                    // pad_amount: 8 DWORDs
  g1[1] = (int)((unsigned)PANEL_COLS << 16);    // tensor_dim0[15:0]
  g1[2] = (int)((unsigned)PANEL_ROWS << 16);    // dim0 hi=0 | tensor_dim1 lo
  g1[3] = (int)((unsigned)PANEL_COLS << 16);    // dim1 hi=0 | tile_dim0
  g1[4] = (int)(PANEL_ROWS & 0xffff);           // tile_dim1 | tile_dim2=0
  g1[5] = (int)stride_elems;                    // tensor_dim0_stride[31:0]
  g1[6] = 0;
  g1[7] = 0;
  i32x4 gz4 = {0, 0, 0, 0};
  i32x8 gz8 = {0, 0, 0, 0, 0, 0, 0, 0};
  __builtin_amdgcn_tensor_load_to_lds(g0, g1, gz4, gz4, gz8, 0);
}
#endif

// Build the orthonormal DCT-II matrix D[k,h] (double-precision cosines).
__global__ __launch_bounds__(256) void dct_init_D(float* __restrict__ Dm) {
  const int idx = blockIdx.x * 256 + threadIdx.x;  // 65536 elements
  const int k = idx >> 8;
  const int h = idx & 255;
  double scale = sqrt(2.0 / 256.0);
  if (k == 0) scale *= (1.0 / sqrt(2.0));
  Dm[idx] = (float)(cos(M_PI * ((double)h + 0.5) * (double)k / 256.0) * scale);
}

// Shared inner chunk: 8 WMMA k-steps from the LDS panel.
#define CHUNK_COMPUTE(BL, AROW, KC)                                         \
  _Pragma("unroll")                                                         \
  for (int kk = 0; kk < PANEL_ROWS; kk += 4) {                              \
    v2f a = *(const v2f*)((AROW) + (KC) + kk);                              \
    const float* Bk = (BL) + kk * PANEL_STRIDE;                             \
    v2f b0 = { Bk[0],  Bk[PANEL_STRIDE + 0]  };                             \
    v2f b1 = { Bk[16], Bk[PANEL_STRIDE + 16] };                             \
    v2f b2 = { Bk[32], Bk[PANEL_STRIDE + 32] };                             \
    v2f b3 = { Bk[48], Bk[PANEL_STRIDE + 48] };                             \
    acc0 = WMMA_F32(a, b0, acc0);                                           \
    acc1 = WMMA_F32(a, b1, acc1);                                           \
    acc2 = WMMA_F32(a, b2, acc2);                                           \
    acc3 = WMMA_F32(a, b3, acc3);                                           \
  }

// Stage 1: per batch, C = D[256x256] @ X_b[256x768]; block = 4 waves covering
// a 64(M) x 64(N) tile; B panel staged via TDM. grid = (128, 4, 12).
__global__ __launch_bounds__(128) void dct_stage1(const float* __restrict__ x,
                                                  const float* __restrict__ Dm,
                                                  float* __restrict__ Tt) {
  __shared__ float bpan[2][PANEL_ROWS * PANEL_STRIDE];

  const int b    = blockIdx.x;
  const int lane = threadIdx.x & 31;
  const int wave = threadIdx.x >> 5;
  const int m0   = (blockIdx.y << 6) + (wave << 4);  // per-wave h' tile
  const int n0   = blockIdx.z << 6;                  // shared (w*3+c) panel
  const int laneLo   = lane & 15;
  const int laneHalf = lane >> 4;

  const float* Arow = Dm + (m0 + laneLo) * 256 + (laneHalf << 1);
  const float* Bb   = x + (size_t)b * (256 * 768) + n0;  // ldb = 768

  v8f acc0 = {}, acc1 = {}, acc2 = {}, acc3 = {};

#if HAVE_TDM
  if (wave == 0)
    tdm_load_2d((unsigned)(size_t)&bpan[0][0], Bb, 768);
#endif
  for (int kc = 0; kc < 256; kc += PANEL_ROWS) {
    const int cur = (kc >> 5) & 1;
#if HAVE_TDM
    if (wave == 0) {
      if (kc + PANEL_ROWS < 256) {
        tdm_load_2d((unsigned)(size_t)&bpan[cur ^ 1][0],
                    Bb + (size_t)(kc + PANEL_ROWS) * 768, 768);
        __builtin_amdgcn_s_wait_tensorcnt(1);  // current chunk done, next in flight
      } else {
        __builtin_amdgcn_s_wait_tensorcnt(0);  // last chunk done
      }
    }
#else
    for (int i = threadIdx.x; i < (PANEL_ROWS * PANEL_COLS) / 4; i += 128) {
      const int r = i >> 4;
      const int cc = (i & 15) << 2;
      *(v4f*)&bpan[cur][r * PANEL_STRIDE + cc] =
          *(const v4f*)(Bb + (size_t)(kc + r) * 768 + cc);
    }
#endif
    __syncthreads();
    __builtin_prefetch(Arow + kc + PANEL_ROWS, 0, 1);  // next A chunk (D, hot)
    const float* Bl = &bpan[cur][(laneHalf << 1) * PANEL_STRIDE + laneLo];
    CHUNK_COMPUTE(Bl, Arow, kc)
    __syncthreads();
  }

  // Transposed store into planar Tt[(b*3+c)][w][h']: per lane the 8 acc values
  // are 8 consecutive h' -> two b128 stores.
  v8f accs[4] = { acc0, acc1, acc2, acc3 };
#pragma unroll
  for (int t = 0; t < 4; ++t) {
    const int n = n0 + (t << 4) + laneLo;  // global column = w*3+c
    const int w = n / 3;
    const int c = n - w * 3;
    float* dst = Tt + (((size_t)(b * 3 + c)) << 16) + (w << 8) + m0 + (laneHalf << 3);
    v4f lo = { accs[t][0], accs[t][1], accs[t][2], accs[t][3] };
    v4f hi = { accs[t][4], accs[t][5], accs[t][6], accs[t][7] };
    *(v4f*)(dst + 0) = lo;
    *(v4f*)(dst + 4) = hi;
  }
}

// Stage 2: per (b,c), Out[k,h'] = D @ Tt_{b,c} (ld=256) + fused epilogue.
// grid = (384, 4, 4), block = 128 (4 waves, 64x64 tile per block).
__global__ __launch_bounds__(128) void dct_stage2(const float* __restrict__ Tt,
                                                  const float* __restrict__ Dm,
                                                  const float* __restrict__ meanp,
                                                  const float* __restrict__ stdp,
                                                  float* __restrict__ out) {
  __shared__ float bpan[2][PANEL_ROWS * PANEL_STRIDE];

  const int bc   = blockIdx.x;   // b*3 + c
  const int b    = bc / 3;
  const int c    = bc - b * 3;
  const int lane = threadIdx.x & 31;
  const int wave = threadIdx.x >> 5;
  const int m0   = (blockIdx.y << 6) + (wave << 4);  // k (W-freq) tile
  const int n0   = blockIdx.z << 6;                  // h' panel
  const int laneLo   = lane & 15;
  const int laneHalf = lane >> 4;

  const float* Arow = Dm + (m0 + laneLo) * 256 + (laneHalf << 1);
  const float* Bb   = Tt + ((size_t)bc << 16) + n0;  // ldb = 256

  v8f acc0 = {}, acc1 = {}, acc2 = {}, acc3 = {};

#if HAVE_TDM
  if (wave == 0)
    tdm_load_2d((unsigned)(size_t)&bpan[0][0], Bb, 256);
#endif
  for (int kc = 0; kc < 256; kc += PANEL_ROWS) {
    const int cur = (kc >> 5) & 1;
#if HAVE_TDM
    if (wave == 0) {
      if (kc + PANEL_ROWS < 256) {
        tdm_load_2d((unsigned)(size_t)&bpan[cur ^ 1][0],
                    Bb + (size_t)(kc + PANEL_ROWS) * 256, 256);
        __builtin_amdgcn_s_wait_tensorcnt(1);
      } else {
        __builtin_amdgcn_s_wait_tensorcnt(0);
      }
    }
#else
    for (int i = threadIdx.x; i < (PANEL_ROWS * PANEL_COLS) / 4; i += 128) {
      const int r = i >> 4;
      const int cc = (i & 15) << 2;
      *(v4f*)&bpan[cur][r * PANEL_STRIDE + cc] =
          *(const v4f*)(Bb + (size_t)(kc + r) * 256 + cc);
    }
#endif
    __syncthreads();
    __builtin_prefetch(Arow + kc + PANEL_ROWS, 0, 1);
    const float* Bl = &bpan[cur][(laneHalf << 1) * PANEL_STRIDE + laneLo];
    CHUNK_COMPUTE(Bl, Arow, kc)
    __syncthreads();
  }

  v8f accs[4] = { acc0, acc1, acc2, acc3 };
#pragma unroll
  for (int t = 0; t < 4; ++t) {
    const int h = n0 + (t << 4) + laneLo;  // h' (H-frequency index)
#pragma unroll
    for (int v = 0; v < 8; ++v) {
      const int k2 = m0 + (laneHalf << 3) + v;   // W-frequency index
      const int mi = ((h << 8) + k2) * 3 + c;    // (H,W,C) flat index
      const float y = accs[t][v];
      const float z = (logf(fabsf(y) + 1e-13f) - meanp[mi]) / stdp[mi];
      out[(size_t)b * 196608 + mi] = z;
    }
  }
}

extern "C" void kernel_launch(void* const* d_in, const int* in_sizes, int n_in,
                              void* d_out, int out_size, void* d_ws, size_t ws_size,
                              hipStream_t stream) {
  (void)in_sizes; (void)n_in; (void)out_size; (void)ws_size;
  const float* x    = (const float*)d_in[0];   // (128,256,256,3) f32
  const float* mean = (const float*)d_in[1];   // (256,256,3) f32
  const float* stdv = (const float*)d_in[2];   // (256,256,3) f32
  float* out = (float*)d_out;                  // (128,256,256,3) f32

  float* Dm = (float*)d_ws;                    // 65536 floats
  float* Tt = Dm + 65536;                      // 128*3*256*256 floats

  dct_init_D<<<256, 256, 0, stream>>>(Dm);
  dct_stage1<<<dim3(128, 4, 12), 128, 0, stream>>>(x, Dm, Tt);
  dct_stage2<<<dim3(384, 4, 4), 128, 0, stream>>>(Tt, Dm, mean, stdv, out);
}